// EMLTreeVectorized_22454089023645
// MI455X (gfx1250) — compile-verified
//
#include <hip/hip_runtime.h>
#include <hip/hip_bf16.h>

#ifndef __has_builtin
#define __has_builtin(x) 0
#endif

#if __has_builtin(__builtin_amdgcn_global_load_async_to_lds_b128)
#define ASYNC_LDS 1
#else
#define ASYNC_LDS 0
#endif

#define NDIM 9
#define NB 256
#define NT 32
#define NV 4

#if ASYNC_LDS
typedef int aml_v4i __attribute__((vector_size(4 * sizeof(int))));
typedef __attribute__((address_space(1))) aml_v4i* aml_gp;   // global src
typedef __attribute__((address_space(3))) aml_v4i* aml_lp;   // LDS dst
// gfx12 CPol: TH in bits[2:0]; TH=1 -> non-temporal (streaming, read-once)
#define AML_CPOL_NT 1
#endif

// dict order from setup_inputs(): x, logits0, noise0, logits1, noise1, ...
struct TreePtrs {
    const float* logits[NDIM];
    const float* noise[NDIM];
};

// LDS float layout:
//   [0      .. 5620)  noise chunks for this (b,t), level l at LOFF[l]
//   [5620   .. 5876)  prev buffer A (256 floats)
//   [5876   .. 6132)  prev buffer B (256 floats)
#define LDS_FLOATS 6132
#define PREV_A 5620
#define PREV_B 5876

__global__ __launch_bounds__(256) void eml_tree_kernel(
    const float* __restrict__ x, TreePtrs p, float* __restrict__ out)
{
    constexpr int NODES[NDIM] = {256, 128, 64, 32, 16, 8, 4, 2, 1};
    constexpr int LOFF[NDIM]  = {0, 2560, 4096, 4864, 5248, 5440, 5536, 5584, 5608};

    __shared__ __align__(16) float lds[LDS_FLOATS];

    const int tid = threadIdx.x;
    const int bt  = blockIdx.x;          // bt = b*T + t
    const int b   = bt / NT;
    const int t   = bt % NT;

    // ---- Stage this (b,t)'s noise (all 9 levels, ~22.5 KB) into LDS ----
    // Each level's chunk is contiguous and 16B aligned in global memory.
    // Issue everything up front on ASYNCcnt for maximal memory-level
    // parallelism; NT policy keeps the 184MB read-once stream out of L2.
#pragma unroll
    for (int l = 0; l < NDIM; ++l) {
        const int c2 = (l == 0) ? 10 : 12;            // C*2 floats per node
        const int n4 = NODES[l] * c2 / 4;             // float4 count
        const float4* g =
            (const float4*)(p.noise[l] + (size_t)bt * (size_t)(NODES[l] * c2));
        float4* d = (float4*)(lds + LOFF[l]);
        for (int j = tid; j < n4; j += 256) {
#if ASYNC_LDS
            __builtin_amdgcn_global_load_async_to_lds_b128(
                (aml_gp)(g + j), (aml_lp)(d + j), 0, AML_CPOL_NT);
#else
            d[j] = g[j];
#endif
        }
    }
#if ASYNC_LDS
#if __has_builtin(__builtin_amdgcn_s_wait_asynccnt)
    __builtin_amdgcn_s_wait_asynccnt(0);
#else
    asm volatile("s_wait_asynccnt 0" ::: "memory");
#endif
#endif
    __syncthreads();

    // x for this batch row: one 16B-aligned vector load (L2-resident)
    const float4 xv4 = *(const float4*)(x + b * NV);
    const float xr[NV] = {xv4.x, xv4.y, xv4.z, xv4.w};

    float* pin  = lds + PREV_A;
    float* pout = lds + PREV_B;

#pragma unroll
    for (int l = 0; l < NDIM; ++l) {
        const int n = NODES[l];
        const int C = (l == 0) ? 5 : 6;
        if (tid < n) {
            const float2* nz2 =
                (const float2*)(lds + LOFF[l] + tid * C * 2);   // 8B aligned
            const float* lg = p.logits[l] + (size_t)(t * n + tid) * (size_t)(C * 2);

            float clast = 0.f, crast = 0.f;
            if (l > 0) {                       // prev[:, :, 0::2] / 1::2
                clast = pin[2 * tid];
                crast = pin[2 * tid + 1];
            }

            // logits row: (c, side) pairs; C=6 rows are 48B/16B-aligned,
            // C=5 rows are 40B/8B-aligned.
            float gl[6], gr[6];
            if (C == 6) {
                const float4* lg4 = (const float4*)lg;
                const float4 a = lg4[0], bq = lg4[1], cq = lg4[2];
                gl[0] = a.x;  gr[0] = a.y;
                gl[1] = a.z;  gr[1] = a.w;
                gl[2] = bq.x; gr[2] = bq.y;
                gl[3] = bq.z; gr[3] = bq.w;
                gl[4] = cq.x; gr[4] = cq.y;
                gl[5] = cq.z; gr[5] = cq.w;
            } else {
                const float2* lg2 = (const float2*)lg;
#pragma unroll
                for (int c = 0; c < 5; ++c) {
                    const float2 v = lg2[c];
                    gl[c] = v.x; gr[c] = v.y;
                }
                gl[5] = 0.f; gr[5] = 0.f;
            }

            float zl[6], zr[6];
#pragma unroll
            for (int c = 0; c < 6; ++c) {
                if (c < C) {
                    const float2 nv = nz2[c];
                    zl[c] = gl[c] + nv.x;      // TAU == 1
                    zr[c] = gr[c] + nv.y;
                }
            }
            // stable softmax-weighted sums (unnormalized exp, one divide)
            float ml = zl[0], mr = zr[0];
#pragma unroll
            for (int c = 1; c < 6; ++c) {
                if (c < C) { ml = fmaxf(ml, zl[c]); mr = fmaxf(mr, zr[c]); }
            }
            float suml = 0.f, sumr = 0.f, dl = 0.f, dr = 0.f;
#pragma unroll
            for (int c = 0; c < 6; ++c) {
                if (c < C) {
                    const float candl = (c == 0) ? 1.f : ((c <= 4) ? xr[c - 1] : clast);
                    const float candr = (c == 0) ? 1.f : ((c <= 4) ? xr[c - 1] : crast);
                    const float el = __expf(zl[c] - ml);
                    const float er = __expf(zr[c] - mr);
                    suml += el;  dl += el * candl;
                    sumr += er;  dr += er * candr;
                }
            }
            float ex = dl / suml;
            ex = fminf(fmaxf(ex, -20.f), 20.f);
            float ey = fmaxf(dr / sumr, 1e-30f);

            // _eml(ex, ey): exp_series(ex + ln_series(ey))
            const float u = ey - 1.f;
            float s = 0.f;
#pragma unroll
            for (int k = 16; k >= 1; --k) {    // ln series, Horner
                const float ck = ((k & 1) ? 1.f : -1.f) / (float)k;
                s = ck + u * s;
            }
            const float tt = ex + u * s;
            float e = 1.f;
#pragma unroll
            for (int k = 12; k >= 1; --k) {    // exp series, Horner
                e = 1.f + tt * e * (1.f / (float)k);
            }
            pout[tid] = e;
        }
        __syncthreads();
        float* tmp = pin; pin = pout; pout = tmp;
    }

    if (tid == 0) __builtin_nontemporal_store(pin[0], out + bt);
}

extern "C" void kernel_launch(void* const* d_in, const int* in_sizes, int n_in,
                              void* d_out, int out_size, void* d_ws, size_t ws_size,
                              hipStream_t stream) {
    (void)in_sizes; (void)n_in; (void)out_size; (void)d_ws; (void)ws_size;
    const float* x = (const float*)d_in[0];
    TreePtrs p;
    for (int l = 0; l < NDIM; ++l) {
        p.logits[l] = (const float*)d_in[1 + 2 * l];   // logits{l}
        p.noise[l]  = (const float*)d_in[2 + 2 * l];   // noise{l}
    }
    float* out = (float*)d_out;
    eml_tree_kernel<<<NB * NT, 256, 0, stream>>>(x, p, out);
}